// MTLAGL_58265526337825
// MI455X (gfx1250) — compile-verified
//
#include <hip/hip_runtime.h>
#include <math.h>

// Problem constants (match reference)
#define NN   10000
#define EE   320000
#define DD   256
#define HH   256
#define LL   3
#define GG   512
#define NGC  10
#define NNC  40
#define PEE  50000
#define KDIM 256   // every GEMM in this network has K == 256

typedef float v2f __attribute__((ext_vector_type(2)));
typedef float v8f __attribute__((ext_vector_type(8)));

// ---------------------------------------------------------------------------
// Small utility kernels
// ---------------------------------------------------------------------------
__global__ __launch_bounds__(256) void k_fill(float* p, float v, int n) {
  int i = blockIdx.x * blockDim.x + threadIdx.x;
  if (i < n) p[i] = v;
}

__global__ __launch_bounds__(256) void k_degree(const int* dst, float* deg, int e) {
  int i = blockIdx.x * blockDim.x + threadIdx.x;
  if (i < e) atomicAdd(&deg[dst[i]], 1.0f);
}

__global__ __launch_bounds__(256) void k_norms(const float* deg, float* inv_sqrt,
                                               float* self_norm, int n) {
  int i = blockIdx.x * blockDim.x + threadIdx.x;
  if (i < n) {
    float d = deg[i] + 1.0f;       // self-loop
    inv_sqrt[i]  = rsqrtf(d);
    self_norm[i] = 1.0f / d;
  }
}

// agg[i,f] = h[i,f] * self_norm[i]   (H == 256, so row = i >> 8)
__global__ __launch_bounds__(256) void k_scale_rows(const float* h, const float* self_norm,
                                                    float* agg, int total) {
  int i = blockIdx.x * blockDim.x + threadIdx.x;
  if (i < total) agg[i] = h[i] * self_norm[i >> 8];
}

// agg[dst] += h[src] * inv_sqrt[src]*inv_sqrt[dst]; one wave per edge
__global__ __launch_bounds__(256) void k_scatter(const float* __restrict__ h,
                                                 const int* __restrict__ src,
                                                 const int* __restrict__ dst,
                                                 const float* __restrict__ inv_sqrt,
                                                 float* __restrict__ agg, int e) {
  int w = blockIdx.x * 8 + (threadIdx.x >> 5);
  if (w >= e) return;
  int lane = threadIdx.x & 31;
  int s = src[w], d = dst[w];
  float norm = inv_sqrt[s] * inv_sqrt[d];
  const float* hp = h + (size_t)s * HH;
  float* ap = agg + (size_t)d * HH;
#pragma unroll
  for (int r = 0; r < HH / 32; ++r) {
    int f = lane + 32 * r;
    atomicAdd(&ap[f], hp[f] * norm);
  }
}

// pooled[batch[i]] += g[i]; one wave per node
__global__ __launch_bounds__(256) void k_pool(const float* __restrict__ g,
                                              const int* __restrict__ batch,
                                              float* __restrict__ pooled, int n) {
  int w = blockIdx.x * 8 + (threadIdx.x >> 5);
  if (w >= n) return;
  int lane = threadIdx.x & 31;
  int b = batch[w];
  const float* gp = g + (size_t)w * HH;
  float* pp = pooled + (size_t)b * HH;
#pragma unroll
  for (int r = 0; r < HH / 32; ++r) {
    int f = lane + 32 * r;
    atomicAdd(&pp[f], gp[f]);
  }
}

// ---------------------------------------------------------------------------
// WMMA fp32 GEMM (K fixed at 256):
//   C[M,Nc] = act( sum_t A_t[M,256] @ W_t[256,Nc] + b0 (+ b1) )
//
// Block = 8 waves computing a 128x16 strip of C. The 256x16 B panel for each
// term is staged in LDS. When Nc is a multiple of 16 (all large GEMMs here)
// the panel rows are contiguous 64B runs in W, so the fill uses the CDNA5
// async global->LDS data mover path (GLOBAL_LOAD_ASYNC_TO_LDS_B128, ASYNCcnt)
// with no VGPR round-trip; the ragged heads (Nc=10/40) use a zero-padded VALU
// fill. The inner loop is completely mask-free:
//   global_load_b64 (A frag) + ds_load_2addr_b32 (B frag) + v_wmma per term.
//
// WMMA f32 16x16x4 lane layouts (CDNA5 ISA 7.12.2):
//   A 16x4 : lane l, vgpr j -> M = l%16, K = 2*(l/16)+j
//   B 4x16 : lane l, vgpr j -> N = l%16, K = 2*(l/16)+j
//   C/D    : lane l, vgpr v -> N = l%16, M = v + 8*(l/16)
// ---------------------------------------------------------------------------
template <int NT, bool RELU>
__global__ __launch_bounds__(256) void k_gemm(
    const float* __restrict__ A0, const float* __restrict__ A1, const float* __restrict__ A2,
    const float* __restrict__ W0, const float* __restrict__ W1, const float* __restrict__ W2,
    const float* __restrict__ b0, const float* __restrict__ b1,
    float* __restrict__ C, int M, int Nc) {
  __shared__ float Bs[NT * KDIM * 16];   // [t][k][16]

  int tiles_n = (Nc + 15) >> 4;
  int block_m = blockIdx.x / tiles_n;
  int tile_n  = blockIdx.x - block_m * tiles_n;

  const float* As[3] = {A0, A1, A2};
  const float* Ws[3] = {W0, W1, W2};

  if ((Nc & 15) == 0) {
    // Async DMA fill: 16 floats per panel row are contiguous in W.
#pragma unroll
    for (int t = 0; t < NT; ++t) {
      const float* __restrict__ W = Ws[t];
      for (int v4 = threadIdx.x; v4 < (KDIM * 16) / 4; v4 += 256) {
        int k  = v4 >> 2;             // panel row
        int cg = (v4 & 3) * 4;        // 4-float group within the 16 columns
        unsigned ldsoff = (unsigned)(size_t)&Bs[t * (KDIM * 16) + k * 16 + cg];
        const float* g = W + (size_t)k * Nc + tile_n * 16 + cg;
        asm volatile("global_load_async_to_lds_b128 %0, %1, off"
                     :: "v"(ldsoff), "v"(g) : "memory");
      }
    }
    asm volatile("s_wait_asynccnt 0x0" ::: "memory");
  } else {
    // Zero-padded VALU fill for ragged output widths (Nc = 10, 40).
#pragma unroll
    for (int t = 0; t < NT; ++t) {
      const float* __restrict__ W = Ws[t];
      for (int idx = threadIdx.x; idx < KDIM * 16; idx += 256) {
        int k = idx >> 4, c = idx & 15;
        int col = tile_n * 16 + c;
        Bs[t * (KDIM * 16) + idx] = (col < Nc) ? W[(size_t)k * Nc + col] : 0.0f;
      }
    }
  }
  __syncthreads();

  int tile_m = block_m * 8 + (threadIdx.x >> 5);
  if (tile_m * 16 >= M) return;

  int lane  = threadIdx.x & 31;
  int lo    = lane & 15;
  int khalf = (lane >> 4) * 2;         // 0 or 2
  int mA    = tile_m * 16 + lo;        // A row for this lane
  int nB    = tile_n * 16 + lo;        // C column for this lane

  v8f acc = {};
#pragma unroll 4
  for (int k0 = 0; k0 < KDIM; k0 += 4) {
#pragma unroll
    for (int t = 0; t < NT; ++t) {
      const float* __restrict__ A = As[t];
      v2f a, b;
      a.x = A[(size_t)mA * KDIM + (k0 + khalf)];
      a.y = A[(size_t)mA * KDIM + (k0 + khalf + 1)];
      b.x = Bs[t * (KDIM * 16) + (k0 + khalf) * 16 + lo];
      b.y = Bs[t * (KDIM * 16) + (k0 + khalf + 1) * 16 + lo];
      // Speculative prefetch of the A stream one cacheline ahead
      // (unconditional: OOB prefetches are silently dropped).
      __builtin_prefetch(&A[(size_t)mA * KDIM + (k0 + 32)], 0, 3);
      acc = __builtin_amdgcn_wmma_f32_16x16x4_f32(
          /*neg_a=*/false, a, /*neg_b=*/false, b,
          /*c_mod=*/(short)0, acc, /*reuse_a=*/false, /*reuse_b=*/false);
    }
  }

  if (nB >= Nc) return;
  float bias = b0[nB];
  if (b1) bias += b1[nB];
#pragma unroll
  for (int v = 0; v < 8; ++v) {
    int mrow = tile_m * 16 + v + 8 * (lane >> 4);
    float val = acc[v] + bias;
    if (RELU) val = fmaxf(val, 0.0f);
    C[(size_t)mrow * Nc + nB] = val;
  }
}

// ---------------------------------------------------------------------------
// Link-prediction folding: p = lpa_w @ lp_w[0:256], q = lpa_w @ lp_w[256:512],
// c = lpa_b . (w0 + w1) + lp_b      (single block, 256 threads)
// ---------------------------------------------------------------------------
__global__ __launch_bounds__(256) void k_lp_vec(const float* __restrict__ lpa_w,
                                                const float* __restrict__ lpa_b,
                                                const float* __restrict__ lp_w,
                                                const float* __restrict__ lp_b,
                                                float* p, float* q, float* cconst) {
  int k = threadIdx.x;
  float sp = 0.0f, sq = 0.0f;
  for (int j = 0; j < HH; ++j) {
    float wv = lpa_w[(size_t)k * HH + j];
    sp += wv * lp_w[j];
    sq += wv * lp_w[HH + j];
  }
  p[k] = sp;
  q[k] = sq;
  if (k == 0) {
    float c = lp_b[0];
    for (int j = 0; j < HH; ++j) c += lpa_b[j] * (lp_w[j] + lp_w[HH + j]);
    cconst[0] = c;
  }
}

// lp[i] = clip(sigmoid(x5[a_i].p + x5[b_i].q + c)); one wave per edge pair
__global__ __launch_bounds__(256) void k_lp(const float* __restrict__ x5,
                                            const int* __restrict__ pos,
                                            const int* __restrict__ neg,
                                            const float* __restrict__ p,
                                            const float* __restrict__ q,
                                            const float* __restrict__ cconst,
                                            float* __restrict__ out) {
  int w = blockIdx.x * 8 + (threadIdx.x >> 5);
  if (w >= 2 * PEE) return;
  int lane = threadIdx.x & 31;
  int a, b;
  if (w < PEE) { a = pos[w];        b = pos[PEE + w]; }
  else         { int i = w - PEE; a = neg[i]; b = neg[PEE + i]; }
  const float* xa = x5 + (size_t)a * HH;
  const float* xb = x5 + (size_t)b * HH;
  float s = 0.0f;
#pragma unroll
  for (int r = 0; r < HH / 32; ++r) {
    int f = lane + 32 * r;
    s += xa[f] * p[f] + xb[f] * q[f];
  }
#pragma unroll
  for (int off = 16; off > 0; off >>= 1) s += __shfl_xor(s, off, 32);
  if (lane == 0) {
    float t = s + cconst[0];
    float sig = 1.0f / (1.0f + __expf(-t));
    sig = fminf(fmaxf(sig, 1e-8f), 1.0f - 1e-8f);
    out[w] = sig;
  }
}

// ---------------------------------------------------------------------------
// Host launcher
// ---------------------------------------------------------------------------
static inline int gemm_blocks(int M, int Nc) {
  int blocks_m = ((M / 16) + 7) / 8;       // 8 tile-rows (128 C rows) per block
  int tiles_n  = (Nc + 15) / 16;
  return blocks_m * tiles_n;
}

extern "C" void kernel_launch(void* const* d_in, const int* in_sizes, int n_in,
                              void* d_out, int out_size, void* d_ws, size_t ws_size,
                              hipStream_t stream) {
  (void)in_sizes; (void)n_in; (void)out_size; (void)ws_size;

  const float* x      = (const float*)d_in[0];
  const int*   edge   = (const int*)d_in[1];   // [2,E] flat: src=edge, dst=edge+E
  const int*   batch  = (const int*)d_in[2];
  const int*   pos    = (const int*)d_in[3];   // [2,PE]
  const int*   neg    = (const int*)d_in[4];   // [2,PE]
  const float* lin1_w = (const float*)d_in[5];
  const float* lin1_b = (const float*)d_in[6];
  const float* gcn_w  = (const float*)d_in[7];   // [L,H,H]
  const float* gcn_b  = (const float*)d_in[8];   // [L,H]
  const float* opl_w  = (const float*)d_in[9];   // [L,H,H]
  const float* opl_b  = (const float*)d_in[10];  // [L,H]
  const float* la_w   = (const float*)d_in[11];  // [3H,H]
  const float* la_b   = (const float*)d_in[12];
  const float* gc1_w  = (const float*)d_in[13];
  const float* gc1_b  = (const float*)d_in[14];
  const float* gc2_w  = (const float*)d_in[15];  // [H,NGC]
  const float* gc2_b  = (const float*)d_in[16];
  const float* nc_w   = (const float*)d_in[17];  // [H,NNC]
  const float* nc_b   = (const float*)d_in[18];
  const float* lpa_w  = (const float*)d_in[19];
  const float* lpa_b  = (const float*)d_in[20];
  const float* lp_w   = (const float*)d_in[21];  // [2H,1]
  const float* lp_b   = (const float*)d_in[22];

  // Workspace layout (floats)
  float* ws    = (float*)d_ws;
  size_t nh    = (size_t)NN * HH;
  float* h     = ws;
  float* agg   = h    + nh;
  float* j0    = agg  + nh;
  float* j1    = j0   + nh;
  float* j2    = j1   + nh;
  float* x5v   = j2   + nh;
  float* gbuf  = x5v  + nh;
  float* pooled= gbuf + nh;                 // G*H
  float* deg   = pooled + (size_t)GG * HH;  // N
  float* invs  = deg  + NN;
  float* selfn = invs + NN;
  float* pvec  = selfn + NN;                // 256
  float* qvec  = pvec + HH;                 // 256
  float* cc    = qvec + HH;                 // 1

  const int* src = edge;
  const int* dst = edge + EE;

  // Degree + normalization
  k_fill<<<(NN + 255) / 256, 256, 0, stream>>>(deg, 0.0f, NN);
  k_degree<<<(EE + 255) / 256, 256, 0, stream>>>(dst, deg, EE);
  k_norms<<<(NN + 255) / 256, 256, 0, stream>>>(deg, invs, selfn, NN);

  // h = x @ lin1_w + lin1_b
  k_gemm<1, false><<<gemm_blocks(NN, HH), 256, 0, stream>>>(
      x, nullptr, nullptr, lin1_w, nullptr, nullptr, lin1_b, nullptr, h, NN, HH);

  // GCN layers: h' = relu(agg @ gcn_w + h @ opl_w + gcn_b + opl_b)
  float* hcur = h;
  float* js[3] = {j0, j1, j2};
  for (int l = 0; l < LL; ++l) {
    k_scale_rows<<<((int)nh + 255) / 256, 256, 0, stream>>>(hcur, selfn, agg, (int)nh);
    k_scatter<<<(EE + 7) / 8, 256, 0, stream>>>(hcur, src, dst, invs, agg, EE);
    k_gemm<2, true><<<gemm_blocks(NN, HH), 256, 0, stream>>>(
        agg, hcur, nullptr,
        gcn_w + (size_t)l * HH * HH, opl_w + (size_t)l * HH * HH, nullptr,
        gcn_b + (size_t)l * HH, opl_b + (size_t)l * HH,
        js[l], NN, HH);
    hcur = js[l];
  }

  // x5 = concat(j0,j1,j2) @ la_w + la_b  (fused 3-term GEMM)
  k_gemm<3, false><<<gemm_blocks(NN, HH), 256, 0, stream>>>(
      j0, j1, j2,
      la_w, la_w + (size_t)HH * HH, la_w + (size_t)2 * HH * HH,
      la_b, nullptr, x5v, NN, HH);

  // g = relu(x5 @ gc1_w + gc1_b)
  k_gemm<1, true><<<gemm_blocks(NN, HH), 256, 0, stream>>>(
      x5v, nullptr, nullptr, gc1_w, nullptr, nullptr, gc1_b, nullptr, gbuf, NN, HH);

  // pooled = segment_sum(g, batch)
  k_fill<<<((GG * HH) + 255) / 256, 256, 0, stream>>>(pooled, 0.0f, GG * HH);
  k_pool<<<(NN + 7) / 8, 256, 0, stream>>>(gbuf, batch, pooled, NN);

  float* out = (float*)d_out;
  float* gc_out = out;                       // [G,NGC]
  float* nc_out = out + (size_t)GG * NGC;    // [N,NNC]
  float* lp_out = nc_out + (size_t)NN * NNC; // [2*PE,1]

  // gc_out = pooled @ gc2_w + gc2_b
  k_gemm<1, false><<<gemm_blocks(GG, NGC), 256, 0, stream>>>(
      pooled, nullptr, nullptr, gc2_w, nullptr, nullptr, gc2_b, nullptr, gc_out, GG, NGC);

  // nc_out = x5 @ nc_w + nc_b
  k_gemm<1, false><<<gemm_blocks(NN, NNC), 256, 0, stream>>>(
      x5v, nullptr, nullptr, nc_w, nullptr, nullptr, nc_b, nullptr, nc_out, NN, NNC);

  // Link prediction: folded weights, then gathered dot products
  k_lp_vec<<<1, 256, 0, stream>>>(lpa_w, lpa_b, lp_w, lp_b, pvec, qvec, cc);
  k_lp<<<(2 * PEE + 7) / 8, 256, 0, stream>>>(x5v, pos, neg, pvec, qvec, cc, lp_out);
}